// PointNet_57535381897223
// MI455X (gfx1250) — compile-verified
//
#include <hip/hip_runtime.h>

// ---------------------------------------------------------------------------
// Types for the fp32 WMMA path: V_WMMA_F32_16X16X4_F32
//   A: 16x4 f32  -> 2 VGPRs/lane, B: 4x16 f32 -> 2 VGPRs/lane, C/D: 16x16 -> 8
// ---------------------------------------------------------------------------
typedef __attribute__((ext_vector_type(2))) float v2f;
typedef __attribute__((ext_vector_type(8))) float v8f;

__device__ __forceinline__ v8f wmma4(v2f a, v2f b, v8f c) {
  // 8 args: (neg_a, A, neg_b, B, c_mod, C, reuse_a, reuse_b)
  return __builtin_amdgcn_wmma_f32_16x16x4_f32(false, a, false, b, (short)0, c,
                                               false, false);
}

// Monotonic float<->uint mapping so float-max == uint-max (handles signs).
__device__ __forceinline__ unsigned fenc(float f) {
  unsigned u = __float_as_uint(f);
  return (u & 0x80000000u) ? ~u : (u | 0x80000000u);
}
__device__ __forceinline__ float fdec(unsigned u) {
  unsigned v = (u & 0x80000000u) ? (u & 0x7fffffffu) : ~u;
  return __uint_as_float(v);
}
#define INIT_ENC 0x007FFFFFu /* fenc(-inf) */

// ---------------------------------------------------------------------------
// Small elementwise kernels
// ---------------------------------------------------------------------------
__global__ void fill_u32_kernel(unsigned* __restrict__ p, long long n, unsigned v) {
  long long i = (long long)blockIdx.x * blockDim.x + threadIdx.x;
  if (i < n) p[i] = v;
}

// decode + (neginf->0 and ReLU collapse to fmaxf(x,0))
__global__ void finalize_relu_kernel(const unsigned* __restrict__ u,
                                     float* __restrict__ o, long long n) {
  long long i = (long long)blockIdx.x * blockDim.x + threadIdx.x;
  if (i < n) o[i] = fmaxf(fdec(u[i]), 0.0f);
}

// global max-pool over nodes into [B,32] encoded accumulators
__global__ void pool_max_kernel(const unsigned* __restrict__ u,
                                const int* __restrict__ batch,
                                unsigned* __restrict__ pool, long long n32) {
  long long i = (long long)blockIdx.x * blockDim.x + threadIdx.x;
  if (i < n32) {
    long long node = i >> 5;
    int c = (int)(i & 31);
    float v = fmaxf(fdec(u[i]), 0.0f); // layer-2 output after relu/neginf fix
    atomicMax(&pool[(long long)batch[node] * 32 + c], fenc(v));
  }
}

// [B,32] @ Wc^T + bc -> [B,C]
__global__ void classifier_kernel(const unsigned* __restrict__ pool,
                                  const float* __restrict__ Wc,
                                  const float* __restrict__ bc,
                                  float* __restrict__ out, int Bn, int C) {
  int idx = blockIdx.x * blockDim.x + threadIdx.x;
  if (idx < Bn * C) {
    int b = idx / C, c = idx % C;
    float s = bc[c];
#pragma unroll
    for (int k = 0; k < 32; ++k)
      s += fmaxf(fdec(pool[b * 32 + k]), 0.0f) * Wc[c * 32 + k];
    out[idx] = s;
  }
}

// ---------------------------------------------------------------------------
// Edge-parallel PointNet layer:
//   per wave: 16 edges. feat[16xKIN] -> relu(GEMM1 + b1) -> GEMM2 + b2
//   -> encoded atomicMax scatter into out_u[dst*32 + ch].
// KP = KIN rounded up to multiple of 4 (zero padded). USE_H: gather h[src].
// ---------------------------------------------------------------------------
template <int KP, int KIN, bool USE_H>
__global__ __launch_bounds__(256) void edge_layer_kernel(
    const float* __restrict__ pos, const int* __restrict__ ei,
    const float* __restrict__ h_in, const float* __restrict__ W1,
    const float* __restrict__ b1, const float* __restrict__ W2,
    const float* __restrict__ b2, unsigned* __restrict__ out_u, long long E_) {
  __shared__ float sW1[KP][32];   // sW1[k][n] = W1[n][k] (B-matrix layout)
  __shared__ float sW2[32][32];
  __shared__ float sB1[32];
  __shared__ float sB2[32];
  __shared__ float feat[8][16][KP + 1];   // per-wave 16xKP feature tile (padded)
  __shared__ float ytile[8][16][33];      // per-wave 16x32 intermediate (padded)
  __shared__ int sdst[8][16];

  const int tid = threadIdx.x;
  for (int i = tid; i < KP * 32; i += 256) {
    int k = i >> 5, n = i & 31;
    sW1[k][n] = (k < KIN) ? W1[n * KIN + k] : 0.0f;
  }
  for (int i = tid; i < 32 * 32; i += 256) {
    int k = i >> 5, n = i & 31;
    sW2[k][n] = W2[n * 32 + k];
  }
  if (tid < 32) {
    sB1[tid] = b1[tid];
    sB2[tid] = b2[tid];
  }
  __syncthreads();

  const int wave = tid >> 5;  // 8 wave32 per block
  const int lane = tid & 31;
  const int m = lane & 15;    // edge row within tile / N within half
  const int half = lane >> 4; // lane half selects K / M sub-tiles

  const long long ebase = ((long long)blockIdx.x * 8 + wave) * 16;
  if (ebase >= E_) return;  // wave-uniform; keeps EXEC all-ones for WMMA

  // ---- gather features for this wave's 16 edges into LDS ----
  {
    long long e = ebase + m;
    long long ec = (e < E_) ? e : (E_ - 1);
    int src = ei[ec];
    int dst = ei[E_ + ec];
    if (half == 0) sdst[wave][m] = dst;
    if (!USE_H) {
      if (half == 0) {
        float px = pos[(long long)src * 3 + 0];
        float py = pos[(long long)src * 3 + 1];
        float pz = pos[(long long)src * 3 + 2];
        feat[wave][m][0] = px;
        feat[wave][m][1] = py;
        feat[wave][m][2] = pz;
        feat[wave][m][3] = px - pos[(long long)dst * 3 + 0];
        feat[wave][m][4] = py - pos[(long long)dst * 3 + 1];
        feat[wave][m][5] = pz - pos[(long long)dst * 3 + 2];
        feat[wave][m][6] = 0.0f;
        feat[wave][m][7] = 0.0f;
      }
    } else {
      // both halves cooperate: half h loads channels [16h, 16h+16)
      const float4* hr =
          reinterpret_cast<const float4*>(h_in + (long long)src * 32) + half * 4;
#pragma unroll
      for (int j = 0; j < 4; ++j) {
        float4 v = hr[j];
        feat[wave][m][half * 16 + 4 * j + 0] = v.x;
        feat[wave][m][half * 16 + 4 * j + 1] = v.y;
        feat[wave][m][half * 16 + 4 * j + 2] = v.z;
        feat[wave][m][half * 16 + 4 * j + 3] = v.w;
      }
      if (half == 0) {
        feat[wave][m][32] = pos[(long long)src * 3 + 0] - pos[(long long)dst * 3 + 0];
        feat[wave][m][33] = pos[(long long)src * 3 + 1] - pos[(long long)dst * 3 + 1];
        feat[wave][m][34] = pos[(long long)src * 3 + 2] - pos[(long long)dst * 3 + 2];
        feat[wave][m][35] = 0.0f;
      }
    }
  }
  __builtin_amdgcn_wave_barrier();  // order LDS writes vs cross-lane reads

  // ---- GEMM1: [16xKP] @ sW1[KPx32] via f32 WMMA 16x16x4 ----
  v8f c0 = {};  // output channels 0..15
  v8f c1 = {};  // output channels 16..31
#pragma unroll
  for (int k = 0; k < KP; k += 4) {
    // A frag: VGPR0 = K{k+0 | k+2}, VGPR1 = K{k+1 | k+3} per lane-half
    v2f a = {feat[wave][m][k + 2 * half], feat[wave][m][k + 2 * half + 1]};
    v2f bA = {sW1[k + 2 * half][m], sW1[k + 2 * half + 1][m]};
    v2f bB = {sW1[k + 2 * half][16 + m], sW1[k + 2 * half + 1][16 + m]};
    c0 = wmma4(a, bA, c0);
    c1 = wmma4(a, bB, c1);
  }

  // bias + ReLU -> intermediate tile in LDS (C layout: VGPR r -> M = r + 8*half)
  {
    const float bn0 = sB1[m], bn1 = sB1[16 + m];
#pragma unroll
    for (int r = 0; r < 8; ++r) {
      int row = r + 8 * half;
      ytile[wave][row][m] = fmaxf(c0[r] + bn0, 0.0f);
      ytile[wave][row][16 + m] = fmaxf(c1[r] + bn1, 0.0f);
    }
  }
  __builtin_amdgcn_wave_barrier();

  // ---- GEMM2: [16x32] @ sW2[32x32] ----
  v8f d0 = {};
  v8f d1 = {};
#pragma unroll
  for (int k = 0; k < 32; k += 4) {
    v2f a = {ytile[wave][m][k + 2 * half], ytile[wave][m][k + 2 * half + 1]};
    v2f bA = {sW2[k + 2 * half][m], sW2[k + 2 * half + 1][m]};
    v2f bB = {sW2[k + 2 * half][16 + m], sW2[k + 2 * half + 1][16 + m]};
    d0 = wmma4(a, bA, d0);
    d1 = wmma4(a, bB, d1);
  }

  // ---- bias + scatter-max (encoded uint atomics, no-return form) ----
  {
    const float cb0 = sB2[m], cb1 = sB2[16 + m];
#pragma unroll
    for (int r = 0; r < 8; ++r) {
      int row = r + 8 * half;
      if (ebase + row < E_) {
        long long node = (long long)sdst[wave][row];
        atomicMax(&out_u[node * 32 + m], fenc(d0[r] + cb0));
        atomicMax(&out_u[node * 32 + 16 + m], fenc(d1[r] + cb1));
      }
    }
  }
}

// ---------------------------------------------------------------------------
// Host launcher
// ---------------------------------------------------------------------------
extern "C" void kernel_launch(void* const* d_in, const int* in_sizes, int n_in,
                              void* d_out, int out_size, void* d_ws,
                              size_t ws_size, hipStream_t stream) {
  const float* pos = (const float*)d_in[0];
  const int* ei = (const int*)d_in[1];     // [2,E] int32
  const int* batch = (const int*)d_in[2];  // [N] int32 (sorted)
  const float* W11 = (const float*)d_in[3];
  const float* b11 = (const float*)d_in[4];
  const float* W12 = (const float*)d_in[5];
  const float* b12 = (const float*)d_in[6];
  const float* W21 = (const float*)d_in[7];
  const float* b21 = (const float*)d_in[8];
  const float* W22 = (const float*)d_in[9];
  const float* b22 = (const float*)d_in[10];
  const float* Wc = (const float*)d_in[11];
  const float* bc = (const float*)d_in[12];
  float* out = (float*)d_out;

  const long long N_ = in_sizes[0] / 3;
  const long long E_ = in_sizes[1] / 2;
  const int C = in_sizes[12];          // 40
  const int Bn = out_size / C;         // 64
  const long long N32 = N_ * 32;

  // workspace layout
  unsigned* u_buf = (unsigned*)d_ws;                          // N*32 u32 (reused)
  float* h_f = (float*)((char*)d_ws + (size_t)N32 * 4);       // N*32 f32
  unsigned* pool_u = (unsigned*)((char*)d_ws + (size_t)N32 * 8);  // Bn*32 u32

  const int TPB = 256;
  const long long tiles = (E_ + 15) / 16;
  const int edge_blocks = (int)((tiles + 7) / 8);
  const int n32_blocks = (int)((N32 + TPB - 1) / TPB);

  // layer 1
  fill_u32_kernel<<<n32_blocks, TPB, 0, stream>>>(u_buf, N32, INIT_ENC);
  fill_u32_kernel<<<(Bn * 32 + TPB - 1) / TPB, TPB, 0, stream>>>(
      pool_u, (long long)Bn * 32, INIT_ENC);
  edge_layer_kernel<8, 6, false><<<edge_blocks, TPB, 0, stream>>>(
      pos, ei, nullptr, W11, b11, W12, b12, u_buf, E_);
  finalize_relu_kernel<<<n32_blocks, TPB, 0, stream>>>(u_buf, h_f, N32);

  // layer 2
  fill_u32_kernel<<<n32_blocks, TPB, 0, stream>>>(u_buf, N32, INIT_ENC);
  edge_layer_kernel<36, 35, true><<<edge_blocks, TPB, 0, stream>>>(
      pos, ei, h_f, W21, b21, W22, b22, u_buf, E_);

  // pool + classifier
  pool_max_kernel<<<n32_blocks, TPB, 0, stream>>>(u_buf, batch, pool_u, N32);
  classifier_kernel<<<(Bn * C + TPB - 1) / TPB, TPB, 0, stream>>>(
      pool_u, Wc, bc, out, Bn, C);
}